// _ProposalLayer_30545807409835
// MI455X (gfx1250) — compile-verified
//
#include <hip/hip_runtime.h>
#include <hip/hip_bf16.h>

// ---------------------------------------------------------------------------
// ProposalLayer for MI455X (gfx1250, wave32).
// Pipeline: decode+key -> bitonic sort (desc) -> gather top4000 ->
//           IoU suppression-mask (TDM box tile into LDS, WMMA f32 16x16x4 for
//           pairwise area sums, ballot_w32 for mask words) -> sequential scan.
// IoU test is division-free: inter/union > TH  <=>  inter > TH*union (union>0).
// ---------------------------------------------------------------------------

typedef float v2f __attribute__((ext_vector_type(2)));
typedef float v8f __attribute__((ext_vector_type(8)));
typedef unsigned int v4u __attribute__((ext_vector_type(4)));
typedef int v8i __attribute__((ext_vector_type(8)));
typedef int v4i __attribute__((ext_vector_type(4)));

#define PAD_N 65536
#define TOPN 4000
#define POSTN 300
#define NWORDS 125     // 4000 / 32
#define NMS_TH 0.7f

#if defined(__gfx1250__) && __has_builtin(__builtin_amdgcn_ballot_w32)
#define BALLOT32(p) __builtin_amdgcn_ballot_w32(p)
#else
#define BALLOT32(p) ((unsigned)__ballot(p))
#endif

// ---------------------------------------------------------------------------
// Kernel 1: decode proposals + build sort keys (score sortable bits | ~idx).
// ---------------------------------------------------------------------------
__global__ void k_init(const float* __restrict__ scores,
                       const float* __restrict__ deltas,
                       const float* __restrict__ im_info,
                       const float* __restrict__ anchors,
                       unsigned long long* __restrict__ keys,
                       float* __restrict__ boxes,
                       int A, int H, int W) {
  int i = blockIdx.x * blockDim.x + threadIdx.x;
  int b = blockIdx.y;
  if (i >= PAD_N) return;
  int K = A * H * W;
  unsigned long long key = 0ull;
  if (i < K) {
    int a = i % A;
    int hw = i / A;
    int w = hw % W;
    int h = hw / W;
    float shx = (float)w * 16.0f;
    float shy = (float)h * 16.0f;
    float x1 = anchors[a * 4 + 0] + shx;
    float y1 = anchors[a * 4 + 1] + shy;
    float x2 = anchors[a * 4 + 2] + shx;
    float y2 = anchors[a * 4 + 3] + shy;
    float aw = x2 - x1 + 1.0f;
    float ah = y2 - y1 + 1.0f;
    float cx = x1 + 0.5f * aw;
    float cy = y1 + 0.5f * ah;
    int plane = H * W;
    size_t dbase = (((size_t)b * (4 * A) + 4 * a) * H + h) * (size_t)W + w;
    float d0 = deltas[dbase];
    float d1 = deltas[dbase + plane];
    float d2 = deltas[dbase + 2 * (size_t)plane];
    float d3 = deltas[dbase + 3 * (size_t)plane];
    float pcx = d0 * aw + cx;
    float pcy = d1 * ah + cy;
    float pw = expf(d2) * aw;
    float ph = expf(d3) * ah;
    float hiX = im_info[b * 3 + 1] - 1.0f;
    float hiY = im_info[b * 3 + 0] - 1.0f;
    float bx1 = fminf(fmaxf(pcx - 0.5f * pw, 0.0f), hiX);
    float by1 = fminf(fmaxf(pcy - 0.5f * ph, 0.0f), hiY);
    float bx2 = fminf(fmaxf(pcx + 0.5f * pw, 0.0f), hiX);
    float by2 = fminf(fmaxf(pcy + 0.5f * ph, 0.0f), hiY);
    size_t bo = ((size_t)b * K + i) * 4;
    boxes[bo + 0] = bx1;
    boxes[bo + 1] = by1;
    boxes[bo + 2] = bx2;
    boxes[bo + 3] = by2;
    float sc = scores[(((size_t)b * A + a) * H + h) * (size_t)W + w];
    unsigned u = __float_as_uint(sc);
    u = (u & 0x80000000u) ? ~u : (u | 0x80000000u);
    key = ((unsigned long long)u << 32) | (unsigned long long)(0xFFFFFFFFu - (unsigned)i);
  }
  keys[(size_t)b * PAD_N + i] = key;
}

// ---------------------------------------------------------------------------
// Bitonic sort (descending), 64-bit keys. 2048-element chunks in LDS.
// ---------------------------------------------------------------------------
__global__ void __launch_bounds__(1024) k_sort_local_full(unsigned long long* __restrict__ keys) {
  __shared__ unsigned long long s[2048];
  int chunk = blockIdx.x;
  int t = threadIdx.x;
  size_t base = (size_t)chunk * 2048;
  s[t] = keys[base + t];
  s[t + 1024] = keys[base + t + 1024];
  int gbase = (chunk & 31) * 2048;  // position inside this image's 65536 block
  for (int k = 2; k <= 2048; k <<= 1) {
    for (int j = k >> 1; j > 0; j >>= 1) {
      __syncthreads();
      int p = ((t & ~(j - 1)) << 1) | (t & (j - 1));
      int q = p | j;
      bool desc = (((gbase + p) & k) == 0);
      unsigned long long x = s[p], y = s[q];
      bool sw = desc ? (x < y) : (x > y);
      if (sw) { s[p] = y; s[q] = x; }
    }
  }
  __syncthreads();
  keys[base + t] = s[t];
  keys[base + t + 1024] = s[t + 1024];
}

__global__ void k_sort_global(unsigned long long* __restrict__ keys, int k, int j) {
  int t = blockIdx.x * blockDim.x + threadIdx.x;  // B * 32768 threads
  int img = t >> 15;
  int tt = t & 32767;
  int p = ((tt & ~(j - 1)) << 1) | (tt & (j - 1));
  size_t i0 = (size_t)img * PAD_N + p;
  size_t i1 = i0 + j;
  bool desc = ((p & k) == 0);
  unsigned long long x = keys[i0], y = keys[i1];
  bool sw = desc ? (x < y) : (x > y);
  if (sw) { keys[i0] = y; keys[i1] = x; }
}

__global__ void __launch_bounds__(1024) k_sort_local_merge(unsigned long long* __restrict__ keys, int k) {
  __shared__ unsigned long long s[2048];
  int chunk = blockIdx.x;
  int t = threadIdx.x;
  size_t base = (size_t)chunk * 2048;
  s[t] = keys[base + t];
  s[t + 1024] = keys[base + t + 1024];
  int gbase = (chunk & 31) * 2048;
  bool desc = (((gbase) & k) == 0);  // k >= 4096: uniform over chunk
  for (int j = 1024; j > 0; j >>= 1) {
    __syncthreads();
    int p = ((t & ~(j - 1)) << 1) | (t & (j - 1));
    int q = p | j;
    unsigned long long x = s[p], y = s[q];
    bool sw = desc ? (x < y) : (x > y);
    if (sw) { s[p] = y; s[q] = x; }
  }
  __syncthreads();
  keys[base + t] = s[t];
  keys[base + t + 1024] = s[t + 1024];
}

// ---------------------------------------------------------------------------
// Kernel: gather top-4000 boxes in score order.
// ---------------------------------------------------------------------------
__global__ void k_gather(const unsigned long long* __restrict__ keys,
                         const float* __restrict__ boxes,
                         float4* __restrict__ sorted, int K) {
  int r = blockIdx.x * blockDim.x + threadIdx.x;
  int b = blockIdx.y;
  if (r >= TOPN) return;
  unsigned long long key = keys[(size_t)b * PAD_N + r];
  unsigned idx = 0xFFFFFFFFu - (unsigned)(key & 0xFFFFFFFFull);
  const float* s = boxes + ((size_t)b * K + idx) * 4;
  sorted[(size_t)b * TOPN + r] = make_float4(s[0], s[1], s[2], s[3]);
}

// ---------------------------------------------------------------------------
// Kernel: IoU suppression mask. TDM loads the full 4000x4 f32 box tile (64 KB)
// into LDS; each wave owns a 16-row x 32-col tile; pairwise area sums come
// from one V_WMMA_F32_16X16X4_F32 per 16x16 half; ballot_w32 packs the mask.
// ---------------------------------------------------------------------------
__global__ void __launch_bounds__(256) k_mask(const float4* __restrict__ sorted,
                                              unsigned* __restrict__ mask) {
  __shared__ float4 shb[TOPN];  // 64000 bytes
  const int b = blockIdx.y;
  const int tid = threadIdx.x;

#if defined(__gfx1250__) && __has_builtin(__builtin_amdgcn_tensor_load_to_lds)
  if (tid < 32) {  // wave 0 issues one TDM op for the whole 64 KB tile
    unsigned long long gaddr =
        (unsigned long long)(const void*)(sorted + (size_t)b * TOPN);
    unsigned ldsOff = (unsigned)(unsigned long long)(void*)&shb[0];
    v4u g0;
    g0.x = 1u;                                      // count=1, user mode
    g0.y = ldsOff;                                  // lds_addr
    g0.z = (unsigned)(gaddr & 0xFFFFFFFFull);       // global_addr lo
    g0.w = (unsigned)((gaddr >> 32) & 0x01FFFFFFull) | 0x80000000u;  // hi | type=2
    v8i g1;
    g1[0] = 0x00020000;           // data_size = 4 bytes
    g1[1] = (int)(16000u << 16);  // tensor_dim0 = 16000 f32 elements
    g1[2] = 0x00010000;           // tensor_dim1 = 1
    g1[3] = (int)(16000u << 16);  // tile_dim0 = 16000
    g1[4] = 1;                    // tile_dim1 = 1
    g1[5] = 16000;                // tensor_dim0_stride
    g1[6] = (int)(16000u << 16);  // tensor_dim1_stride[15:0]
    g1[7] = 0;
    v4i z4 = {0, 0, 0, 0};
    v8i z8 = {0, 0, 0, 0, 0, 0, 0, 0};
    __builtin_amdgcn_tensor_load_to_lds(g0, g1, z4, z4, z8, 0);
    __builtin_amdgcn_s_wait_tensorcnt(0);
  }
#else
  {
    const float* src = (const float*)(sorted + (size_t)b * TOPN);
    float* dst = (float*)&shb[0];
    for (int i = tid; i < TOPN * 4; i += 256) dst[i] = src[i];
  }
#endif
  __syncthreads();

  const int lane = tid & 31;
  const int wv = tid >> 5;
  const int rt = blockIdx.x * 8 + wv;  // row tile of 16 rows
  if (rt >= TOPN / 16) return;         // wave-uniform exit
  const int rowBase = rt * 16;
  const int mOff = (lane >> 4) << 3;   // lanes 0-15 -> rows k, lanes 16-31 -> rows k+8
  float4 rb[8];
#pragma unroll
  for (int k = 0; k < 8; ++k) rb[k] = shb[rowBase + mOff + k];
  float4 arl = shb[rowBase + (lane & 15)];
  float areaRowLane = (arl.z - arl.x) * (arl.w - arl.y);

  for (int cw = 0; cw < NWORDS; ++cw) {
    unsigned bal[2][8];
#pragma unroll
    for (int half = 0; half < 2; ++half) {
      const int colBase = cw * 32 + half * 16;
      float4 cb = shb[colBase + (lane & 15)];
      float areaCol = (cb.z - cb.x) * (cb.w - cb.y);
      float sum8[8];
#if defined(__gfx1250__) && __has_builtin(__builtin_amdgcn_wmma_f32_16x16x4_f32)
      // D[m][n] = areaRow[m]*1 + 1*areaCol[n]  (rank-2 outer sum via matrix core)
      v2f Aop, Bop;
      Aop.x = (lane < 16) ? areaRowLane : 0.0f;  // A[m][K=0] = areaRow[m]
      Aop.y = (lane < 16) ? 1.0f : 0.0f;         // A[m][K=1] = 1
      Bop.x = (lane < 16) ? 1.0f : 0.0f;         // B[K=0][n] = 1
      Bop.y = (lane < 16) ? areaCol : 0.0f;      // B[K=1][n] = areaCol[n]
      v8f Cz = {0.f, 0.f, 0.f, 0.f, 0.f, 0.f, 0.f, 0.f};
      v8f D = __builtin_amdgcn_wmma_f32_16x16x4_f32(false, Aop, false, Bop,
                                                    (short)0, Cz, false, false);
#pragma unroll
      for (int k = 0; k < 8; ++k) sum8[k] = D[k];
#else
#pragma unroll
      for (int k = 0; k < 8; ++k)
        sum8[k] = (rb[k].z - rb[k].x) * (rb[k].w - rb[k].y) + areaCol;
#endif
      const int col = colBase + (lane & 15);
#pragma unroll
      for (int k = 0; k < 8; ++k) {
        float ix1 = fmaxf(rb[k].x, cb.x);
        float iy1 = fmaxf(rb[k].y, cb.y);
        float ix2 = fminf(rb[k].z, cb.z);
        float iy2 = fminf(rb[k].w, cb.w);
        float iw = fmaxf(ix2 - ix1, 0.0f);
        float ih = fmaxf(iy2 - iy1, 0.0f);
        float inter = iw * ih;
        // inter/(union+eps) > TH  <=>  inter > TH*(union+eps), union+eps > 0
        float uni = sum8[k] - inter + 1e-9f;
        const int row = rowBase + mOff + k;
        bool pred = (inter > NMS_TH * uni) && (col > row);
        bal[half][k] = BALLOT32(pred);
      }
    }
    if (lane == 0) {
      unsigned* mrow = mask + ((size_t)b * TOPN) * NWORDS + cw;
#pragma unroll
      for (int k = 0; k < 8; ++k) {
        // ballot bits[15:0] = row k (cols 0-15), bits[31:16] = row k+8
        unsigned w0 = (bal[0][k] & 0xFFFFu) | ((bal[1][k] & 0xFFFFu) << 16);
        unsigned w1 = (bal[0][k] >> 16) | (bal[1][k] & 0xFFFF0000u);
        mrow[(size_t)(rowBase + k) * NWORDS] = w0;
        mrow[(size_t)(rowBase + k + 8) * NWORDS] = w1;
      }
    }
  }
}

// ---------------------------------------------------------------------------
// Kernel: sequential greedy suppression scan + compaction + output.
// One block per image; 125 lanes each own one mask word of the running remv.
// ---------------------------------------------------------------------------
__global__ void __launch_bounds__(128) k_scan(const unsigned* __restrict__ mask,
                                              const float4* __restrict__ sorted,
                                              float* __restrict__ out) {
  const int b = blockIdx.x;
  const int t = threadIdx.x;
  __shared__ unsigned remv[NWORDS];
  __shared__ unsigned keepw[NWORDS];
  __shared__ int curKeep;
  if (t < NWORDS) { remv[t] = 0u; keepw[t] = 0u; }
  for (int i = t; i < POSTN * 5; i += 128) out[(size_t)b * POSTN * 5 + i] = 0.0f;
  __syncthreads();
  for (int i = 0; i < TOPN; ++i) {
    if (t == 0) curKeep = ((remv[i >> 5] >> (i & 31)) & 1u) ? 0 : 1;
    __syncthreads();
    if (curKeep) {
      if (t < NWORDS) remv[t] |= mask[((size_t)b * TOPN + i) * NWORDS + t];
      if (t == 0) keepw[i >> 5] |= (1u << (i & 31));
    }
    __syncthreads();
  }
  if (t == 0) {
    int rank = 0;
    for (int i = 0; i < TOPN && rank < POSTN; ++i) {
      if ((keepw[i >> 5] >> (i & 31)) & 1u) {
        float4 bx = sorted[(size_t)b * TOPN + i];
        float* o = out + ((size_t)b * POSTN + rank) * 5;
        o[0] = (float)b;
        o[1] = bx.x; o[2] = bx.y; o[3] = bx.z; o[4] = bx.w;
        ++rank;
      }
    }
  }
}

// ---------------------------------------------------------------------------
extern "C" void kernel_launch(void* const* d_in, const int* in_sizes, int n_in,
                              void* d_out, int out_size, void* d_ws, size_t ws_size,
                              hipStream_t stream) {
  const float* scores = (const float*)d_in[0];
  const float* deltas = (const float*)d_in[1];
  const float* im_info = (const float*)d_in[2];
  const float* anchors = (const float*)d_in[3];
  const int B = in_sizes[2] / 3;   // im_info is (B,3)
  const int A = in_sizes[3] / 4;   // anchors is (A,4)
  const int H = 50, W = 76;
  const int K = A * H * W;         // 34200

  char* ws = (char*)d_ws;
  unsigned long long* keys = (unsigned long long*)ws;          // B*65536*8
  size_t off = (size_t)B * PAD_N * 8;
  float* boxes = (float*)(ws + off);                           // B*K*16
  off += (size_t)B * K * 16;
  off = (off + 255) & ~(size_t)255;
  float4* sorted = (float4*)(ws + off);                        // B*4000*16
  off += (size_t)B * TOPN * 16;
  off = (off + 255) & ~(size_t)255;
  unsigned* mask = (unsigned*)(ws + off);                      // B*4000*125*4

  // 1. decode + sort keys
  k_init<<<dim3(PAD_N / 256, B), 256, 0, stream>>>(scores, deltas, im_info,
                                                   anchors, keys, boxes, A, H, W);
  // 2. bitonic sort, descending, per-image over 65536 keys
  k_sort_local_full<<<B * 32, 1024, 0, stream>>>(keys);
  for (int k = 4096; k <= PAD_N; k <<= 1) {
    for (int j = k >> 1; j >= 2048; j >>= 1)
      k_sort_global<<<(B * 32768) / 256, 256, 0, stream>>>(keys, k, j);
    k_sort_local_merge<<<B * 32, 1024, 0, stream>>>(keys, k);
  }
  // 3. gather top-4000 boxes in score order
  k_gather<<<dim3((TOPN + 255) / 256, B), 256, 0, stream>>>(keys, boxes, sorted, K);
  // 4. IoU suppression mask (TDM + WMMA + ballot)
  k_mask<<<dim3(32, B), 256, 0, stream>>>(sorted, mask);
  // 5. sequential suppression scan + emit
  k_scan<<<B, 128, 0, stream>>>(mask, sorted, (float*)d_out);
}